// HypergraphConvLayer_10299331576565
// MI455X (gfx1250) — compile-verified
//
#include <hip/hip_runtime.h>
#include <hip/hip_bf16.h>

// ---------------------------------------------------------------------------
// Hypergraph conv layer for MI455X (gfx1250, wave32).
//   edge_sum = segment-sum of gathered node features      (memory bound)
//   agg      = scatter-add of edge_sum back to nodes      (L2 atomic bound)
//   out      = relu(agg @ weight)  via V_WMMA_F32_16X16X4_F32 (fp32 WMMA)
// ---------------------------------------------------------------------------

#define N_NODES   50000
#define N_EDGES   20000
#define EDGE_SIZE 32
#define IN_CH     256
#define OUT_CH    256

typedef __attribute__((ext_vector_type(2))) float v2f;
typedef __attribute__((ext_vector_type(8))) float v8f;

// ---------------------------------------------------------------- zero agg
__global__ __launch_bounds__(256) void hg_zero_kernel(float4* __restrict__ p,
                                                      unsigned int n4) {
    unsigned int i = blockIdx.x * 256u + threadIdx.x;
    unsigned int stride = gridDim.x * 256u;
    float4 z; z.x = 0.f; z.y = 0.f; z.z = 0.f; z.w = 0.f;
    for (; i < n4; i += stride) p[i] = z;
}

// --------------------------------------------------- per-edge gather + sum
// block = 256 threads (one channel each), one block per hyperedge
__global__ __launch_bounds__(256) void hg_edge_sum_kernel(
        const float* __restrict__ x, const int* __restrict__ he,
        float* __restrict__ edge_sum) {
    __shared__ int idx[EDGE_SIZE];
    const int e = blockIdx.x;
    const int c = threadIdx.x;
    if (c < EDGE_SIZE) idx[c] = he[e * EDGE_SIZE + c];
    __syncthreads();
    float acc = 0.f;
#pragma unroll
    for (int k = 0; k < EDGE_SIZE; ++k)
        acc += x[(size_t)idx[k] * IN_CH + c];
    edge_sum[(size_t)e * IN_CH + c] = acc;
}

// ------------------------------------------------------- per-edge scatter
// block = 256 threads (one channel each); fp32 atomics into agg rows.
__global__ __launch_bounds__(256) void hg_scatter_kernel(
        const int* __restrict__ he, const float* __restrict__ edge_sum,
        float* __restrict__ agg) {
    __shared__ int idx[EDGE_SIZE];
    const int e = blockIdx.x;
    const int c = threadIdx.x;
    if (c < EDGE_SIZE) idx[c] = he[e * EDGE_SIZE + c];
    __syncthreads();
    const float v = edge_sum[(size_t)e * IN_CH + c];
#pragma unroll
    for (int k = 0; k < EDGE_SIZE; ++k) {
        float* dst = &agg[(size_t)idx[k] * IN_CH + c];
        __hip_atomic_fetch_add(dst, v, __ATOMIC_RELAXED,
                               __HIP_MEMORY_SCOPE_AGENT);
    }
}

// -------------------------------------------- fp32 WMMA GEMM + ReLU epilog
// grid = (ceil(3125/8), 16); block = 256 threads = 8 wave32 waves.
// Each block: stages weight panel W[:, nTile*16 .. +16) (256x16 f32, 16 KB)
// into LDS. Each wave computes one 16(M) x 16(N) output tile, K-loop step 4.
//
// fp32 A-frag (16x4) lane layout (ISA 7.12.2):
//   lane L: M = L%16 ; VGPR0 = A[M][klo], VGPR1 = A[M][klo+1], klo = (L/16)*2
// fp32 B-frag (4x16) mirrors it: lane L: N = L%16 ; rows klo, klo+1.
// C/D (16x16 f32, 8 VGPRs): VGPR r -> M = r (+8 for lanes 16..31), N = L%16.
__global__ __launch_bounds__(256) void hg_gemm_relu_kernel(
        const float* __restrict__ A,   // agg [N_NODES, IN_CH]
        const float* __restrict__ W,   // weight [IN_CH, OUT_CH]
        float* __restrict__ out) {
    __shared__ float wpanel[IN_CH * 16];   // [k][n], n = 0..15

    const int tid     = threadIdx.x;
    const int nTile   = blockIdx.y;
    const int colBase = nTile * 16;

    // Stage the 256x16 weight panel into LDS.
#pragma unroll
    for (int i = tid; i < IN_CH * 16; i += 256) {
        const int k = i >> 4;
        const int n = i & 15;
        wpanel[i] = W[k * OUT_CH + colBase + n];
    }
    __syncthreads();

    const int wave  = tid >> 5;           // 0..7
    const int lane  = tid & 31;           // wave32
    const int mTile = blockIdx.x * 8 + wave;
    const int numMTiles = N_NODES / 16;   // 3125 (exact)
    if (mTile >= numMTiles) return;       // whole-wave skip: EXEC stays all-1s

    const int rowBase = mTile * 16;
    const int mn      = lane & 15;        // M for A-frag, N for B-frag
    const int klo     = (lane >> 4) << 1; // 0 or 2

    const float* arow = A + (size_t)(rowBase + mn) * IN_CH + klo;

    v8f acc = {};
#pragma unroll 8
    for (int k = 0; k < IN_CH; k += 4) {
        // Prefetch the A cacheline ~8 WMMA steps ahead (global_prefetch_b8).
        if ((k & 31) == 0)
            __builtin_prefetch(arow + k + 32, 0, 1);

        v2f a = *(const v2f*)(arow + k);                 // A[M][k+klo .. +1]
        v2f b;
        b.x = wpanel[(k + klo)     * 16 + mn];           // B[k+klo  ][N]
        b.y = wpanel[(k + klo + 1) * 16 + mn];           // B[k+klo+1][N]

        // D = A x B + C   (v_wmma_f32_16x16x4_f32)
        acc = __builtin_amdgcn_wmma_f32_16x16x4_f32(
                /*neg_a=*/false, a, /*neg_b=*/false, b,
                /*c_mod=*/(short)0, acc,
                /*reuse_a=*/false, /*reuse_b=*/false);
    }

    // ReLU + store: VGPR r holds (M = r or r+8, N = lane%16).
    const int  mOff = (lane < 16) ? 0 : 8;
    float* orow = out + (size_t)rowBase * OUT_CH + colBase + mn;
#pragma unroll
    for (int r = 0; r < 8; ++r) {
        float v = acc[r];
        v = v > 0.f ? v : 0.f;
        orow[(size_t)(r + mOff) * OUT_CH] = v;
    }
}

// ---------------------------------------------------------------------------
extern "C" void kernel_launch(void* const* d_in, const int* in_sizes, int n_in,
                              void* d_out, int out_size, void* d_ws, size_t ws_size,
                              hipStream_t stream) {
    const float* x  = (const float*)d_in[0];   // [N_NODES, IN_CH]
    const int*   he = (const int*)d_in[1];     // [N_EDGES, EDGE_SIZE]
    const float* w  = (const float*)d_in[2];   // [IN_CH, OUT_CH]
    float*       out = (float*)d_out;          // [N_NODES, OUT_CH]

    // Workspace layout: edge_sum (20.48 MB) | agg (51.2 MB)
    float* edge_sum = (float*)d_ws;
    float* agg = (float*)((char*)d_ws + (size_t)N_EDGES * IN_CH * sizeof(float));

    const unsigned int agg_f4 = (unsigned int)((size_t)N_NODES * IN_CH / 4);
    hg_zero_kernel<<<2048, 256, 0, stream>>>((float4*)agg, agg_f4);

    hg_edge_sum_kernel<<<N_EDGES, 256, 0, stream>>>(x, he, edge_sum);

    hg_scatter_kernel<<<N_EDGES, 256, 0, stream>>>(he, edge_sum, agg);

    dim3 grid((N_NODES / 16 + 7) / 8, OUT_CH / 16);   // (391, 16)
    hg_gemm_relu_kernel<<<grid, 256, 0, stream>>>(agg, w, out);
}